// CommAwareGAT_23106924053104
// MI455X (gfx1250) — compile-verified
//
#include <hip/hip_runtime.h>

typedef __attribute__((ext_vector_type(2))) float v2f;
typedef __attribute__((ext_vector_type(8))) float v8f;

#define IN_CH  128
#define OUT_CH 64

// -------------------------------------------------------------------------
// Kernel 1: h = x @ W1 via V_WMMA_F32_16X16X4_F32.
// One wave computes a 16(row) x 64(col) tile of h.
//   A tile (16x4 f32): lanes 0-15 -> M=lane, K={k,k+1}; lanes 16-31 -> M=lane-16, K={k+2,k+3}
//   B tile (4x16 f32): lanes 0-15 -> N=lane, K={k,k+1}; lanes 16-31 -> N=lane-16, K={k+2,k+3}
//   C/D (16x16 f32):   VGPR j: lanes 0-15 -> M=j, N=lane; lanes 16-31 -> M=j+8, N=lane-16
// Out-of-range rows are CLAMPED for loads (always legal address, no exec
// divergence in the hot loop) and masked only at the store.
// -------------------------------------------------------------------------
__global__ void __launch_bounds__(256)
gat_gemm_wmma(const float* __restrict__ x, const float* __restrict__ W1,
              float* __restrict__ h, int n_nodes) {
  const int wave = (blockIdx.x * blockDim.x + threadIdx.x) >> 5;
  const int lane = threadIdx.x & 31;
  const int m0   = wave * 16;
  if (m0 >= n_nodes) return;

  const int half = lane >> 4;       // 0 = lanes 0-15, 1 = lanes 16-31
  const int l16  = lane & 15;
  int rowA = m0 + l16;              // A-matrix row held by this lane
  if (rowA > n_nodes - 1) rowA = n_nodes - 1;   // clamp: loads always in-bounds

  v8f acc0 = {}, acc1 = {}, acc2 = {}, acc3 = {};

  const float* __restrict__ xrow = x + (size_t)rowA * IN_CH + half * 2;

#pragma unroll
  for (int k = 0; k < IN_CH; k += 4) {
    // A K-pair: one 8-byte vector load (kk even, row base 512B-aligned)
    const v2f a = *(const v2f*)(xrow + k);

    const int kk = k + half * 2;
    const float* __restrict__ b0 = W1 + (size_t)kk * OUT_CH + l16;
    const float* __restrict__ b1 = b0 + OUT_CH;

    v2f bt;
    // N-tile 0 (cols 0-15)
    bt.x = b0[0];  bt.y = b1[0];
    acc0 = __builtin_amdgcn_wmma_f32_16x16x4_f32(false, a, false, bt, (short)0, acc0, false, false);
    // N-tile 1 (cols 16-31)
    bt.x = b0[16]; bt.y = b1[16];
    acc1 = __builtin_amdgcn_wmma_f32_16x16x4_f32(false, a, false, bt, (short)0, acc1, false, false);
    // N-tile 2 (cols 32-47)
    bt.x = b0[32]; bt.y = b1[32];
    acc2 = __builtin_amdgcn_wmma_f32_16x16x4_f32(false, a, false, bt, (short)0, acc2, false, false);
    // N-tile 3 (cols 48-63)
    bt.x = b0[48]; bt.y = b1[48];
    acc3 = __builtin_amdgcn_wmma_f32_16x16x4_f32(false, a, false, bt, (short)0, acc3, false, false);
  }

  // Store D: row = m0 + j + half*8, col = tile*16 + l16
#pragma unroll
  for (int j = 0; j < 8; ++j) {
    const int row = m0 + j + half * 8;
    if (row < n_nodes) {
      float* __restrict__ hrow = h + (size_t)row * OUT_CH + l16;
      hrow[0]  = acc0[j];
      hrow[16] = acc1[j];
      hrow[32] = acc2[j];
      hrow[48] = acc3[j];
    }
  }
}

// -------------------------------------------------------------------------
// Kernel 2: per-node attention scalars.
//   a_dst[n] = h[n,:] . Wm_w[0:64]    (multiplies h_i = h[dst])
//   a_src[n] = h[n,:] . Wm_w[64:128]  (multiplies h_j = h[src])
// One wave32 per node; lanes cover 2 channels each, wave reduction.
// -------------------------------------------------------------------------
__global__ void __launch_bounds__(256)
gat_node_scalars(const float* __restrict__ h, const float* __restrict__ Wm_w,
                 float* __restrict__ a_dst, float* __restrict__ a_src, int n_nodes) {
  const int node = (blockIdx.x * blockDim.x + threadIdx.x) >> 5;
  const int lane = threadIdx.x & 31;
  if (node >= n_nodes) return;
  const float* __restrict__ hn = h + (size_t)node * OUT_CH;
  const float h0 = hn[lane], h1 = hn[lane + 32];
  float d = h0 * Wm_w[lane]      + h1 * Wm_w[lane + 32];
  float s = h0 * Wm_w[lane + 64] + h1 * Wm_w[lane + 96];
#pragma unroll
  for (int off = 16; off > 0; off >>= 1) {
    d += __shfl_down(d, off, 32);
    s += __shfl_down(s, off, 32);
  }
  if (lane == 0) { a_dst[node] = d; a_src[node] = s; }
}

// -------------------------------------------------------------------------
// Kernel 3: init out[n,c] = bias[c]; denom[n] = 0.
// -------------------------------------------------------------------------
__global__ void __launch_bounds__(256)
gat_init(float* __restrict__ out, const float* __restrict__ bias,
         float* __restrict__ denom, int n_nodes) {
  const int i = blockIdx.x * blockDim.x + threadIdx.x;
  if (i < n_nodes * OUT_CH) out[i] = bias[i & (OUT_CH - 1)];
  if (i < n_nodes)          denom[i] = 0.0f;
}

// -------------------------------------------------------------------------
// Kernel 4: edge scores: num[e] = exp(leaky_relu(a_dst[dst]+a_src[src]+b, 0.2));
//           denom[dst] += num[e]   (atomic)
// -------------------------------------------------------------------------
__global__ void __launch_bounds__(256)
gat_edge_scores(const long long* __restrict__ src, const long long* __restrict__ dst,
                const float* __restrict__ a_dst, const float* __restrict__ a_src,
                const float* __restrict__ Wm_b,
                float* __restrict__ num, float* __restrict__ denom, int n_edges) {
  const int e = blockIdx.x * blockDim.x + threadIdx.x;
  if (e >= n_edges) return;
  const int s = (int)src[e];
  const int d = (int)dst[e];
  float sc = a_dst[d] + a_src[s] + Wm_b[0];
  sc = sc > 0.0f ? sc : 0.2f * sc;          // leaky_relu(., 0.2)
  const float nm = expf(sc);
  num[e] = nm;
  atomicAdd(&denom[d], nm);
}

// -------------------------------------------------------------------------
// Kernel 5: alpha[e] = num[e] / (denom[src[e]] + 1e-16)   (reference gathers by src)
// -------------------------------------------------------------------------
__global__ void __launch_bounds__(256)
gat_edge_alpha(const long long* __restrict__ src, const float* __restrict__ denom,
               float* __restrict__ num, int n_edges) {
  const int e = blockIdx.x * blockDim.x + threadIdx.x;
  if (e >= n_edges) return;
  num[e] = num[e] / (denom[(int)src[e]] + 1e-16f);
}

// -------------------------------------------------------------------------
// Kernel 6: out[src[e], c] += h[src[e], c] * alpha[e]  (reference scatters by src)
// 64 threads per edge, one channel each.
// -------------------------------------------------------------------------
__global__ void __launch_bounds__(256)
gat_aggregate(const long long* __restrict__ src, const float* __restrict__ h,
              const float* __restrict__ alpha, float* __restrict__ out, int n_edges) {
  const long long gid = (long long)blockIdx.x * blockDim.x + threadIdx.x;
  const int e = (int)(gid >> 6);
  const int c = (int)(gid & 63);
  if (e >= n_edges) return;
  const int s = (int)src[e];
  atomicAdd(&out[(size_t)s * OUT_CH + c], h[(size_t)s * OUT_CH + c] * alpha[e]);
}

// -------------------------------------------------------------------------
extern "C" void kernel_launch(void* const* d_in, const int* in_sizes, int n_in,
                              void* d_out, int out_size, void* d_ws, size_t ws_size,
                              hipStream_t stream) {
  const float*     x    = (const float*)d_in[0];      // (1, N, 128)
  const long long* ei   = (const long long*)d_in[1];  // (1, 2, E) int64
  const float*     W1   = (const float*)d_in[3];      // (128, 64)
  const float*     Wm_w = (const float*)d_in[4];      // (128, 1)
  const float*     Wm_b = (const float*)d_in[5];      // (1,)
  const float*     bias = (const float*)d_in[6];      // (64,)
  float*           out  = (float*)d_out;              // (1, N, 64)

  const int n_nodes = in_sizes[0] / IN_CH;
  const int n_edges = in_sizes[1] / 2;
  const long long* src = ei;             // edge_index[:,0,:]
  const long long* dst = ei + n_edges;   // edge_index[:,1,:]

  // Workspace layout
  char* ws = (char*)d_ws;
  float* h     = (float*)ws;                         ws += (size_t)n_nodes * OUT_CH * sizeof(float);
  float* a_dst = (float*)ws;                         ws += (size_t)n_nodes * sizeof(float);
  float* a_src = (float*)ws;                         ws += (size_t)n_nodes * sizeof(float);
  float* denom = (float*)ws;                         ws += (size_t)n_nodes * sizeof(float);
  float* num   = (float*)ws;                         // n_edges floats

  // 1) GEMM with WMMA (one wave per 16-row tile)
  {
    const int waves   = (n_nodes + 15) / 16;
    const int threads = waves * 32;
    const int blocks  = (threads + 255) / 256;
    gat_gemm_wmma<<<blocks, 256, 0, stream>>>(x, W1, h, n_nodes);
  }
  // 2) Per-node attention scalars (one wave per node)
  {
    const int blocks = (n_nodes * 32 + 255) / 256;
    gat_node_scalars<<<blocks, 256, 0, stream>>>(h, Wm_w, a_dst, a_src, n_nodes);
  }
  // 3) Init out=bias, denom=0
  {
    const int blocks = (n_nodes * OUT_CH + 255) / 256;
    gat_init<<<blocks, 256, 0, stream>>>(out, bias, denom, n_nodes);
  }
  // 4) Edge scores + denom scatter
  {
    const int blocks = (n_edges + 255) / 256;
    gat_edge_scores<<<blocks, 256, 0, stream>>>(src, dst, a_dst, a_src, Wm_b, num, denom, n_edges);
  }
  // 5) Alpha (in-place over num)
  {
    const int blocks = (n_edges + 255) / 256;
    gat_edge_alpha<<<blocks, 256, 0, stream>>>(src, denom, num, n_edges);
  }
  // 6) Weighted aggregation
  {
    const long long total = (long long)n_edges * OUT_CH;
    const int blocks = (int)((total + 255) / 256);
    gat_aggregate<<<blocks, 256, 0, stream>>>(src, h, num, out, n_edges);
  }
}